// CRF_67267777790051
// MI455X (gfx1250) — compile-verified
//
#include <hip/hip_runtime.h>
#include <hip/hip_bf16.h>

typedef __attribute__((ext_vector_type(2))) float v2f;
typedef __attribute__((ext_vector_type(8))) float v8f;

namespace {
constexpr int kB  = 128;
constexpr int kS  = 2048;
constexpr int kD  = 128;
constexpr int kY  = 26;
constexpr int kYp = 32;   // padded label count for WMMA N-tiles
}

// ---------------------------------------------------------------------------
// Kernel 1: emit[row, y] = sum_d X[row, d] * W[d, y]   (row = b*S + s)
// One wave per 16-row M-tile; K=128 swept in steps of 4 with
// V_WMMA_F32_16X16X4_F32; N=26 covered by two 16-wide N-tiles (cols 26..31 of
// the B matrix are zero-padded in LDS).  Output written unpadded [row][26]
// directly into d_out, which kernel 2 later overwrites with the one-hot path.
//
// Assumed fp32 WMMA operand layouts (ISA 7.12.2):
//   A 16x4 : lanes 0-15 = rows M=0..15 with {K=0,K=1}; lanes 16-31 = {K=2,K=3}
//   B 4x16 : lanes 0-15 = cols N=0..15 with {K=0,K=1}; lanes 16-31 = {K=2,K=3}
//   C/D    : VGPR r -> M=r (lanes 0-15) / M=r+8 (lanes 16-31), N = lane&15
// ---------------------------------------------------------------------------
__global__ __launch_bounds__(256) void crf_emit_wmma(
    const float* __restrict__ X, const float* __restrict__ W,
    float* __restrict__ emit) {
  __shared__ float Wl[kD * kYp];
  const int t = threadIdx.x;
  // Stage W (128x26) into LDS zero-padded to 128x32. Uniform, branch-free.
#pragma unroll
  for (int i = 0; i < (kD * kYp) / 256; ++i) {
    const int idx = t + i * 256;
    const int d = idx >> 5, y = idx & 31;
    Wl[idx] = (y < kY) ? W[d * kY + y] : 0.0f;
  }
  __syncthreads();

  const int lane = t & 31;
  const int wave = t >> 5;
  const int tile = blockIdx.x * 8 + wave;   // 16384 tiles / 8 waves = 2048 blocks
  const int row0 = tile * 16;
  const int hs   = lane >> 4;               // half-select (K pair 0/1 vs 2/3)
  const int l16  = lane & 15;

  const float* arow = X + (size_t)(row0 + l16) * kD;

  v8f c0 = {};   // N-tile cols 0..15
  v8f c1 = {};   // N-tile cols 16..31 (only 16..25 stored)
#pragma unroll 4
  for (int k = 0; k < kD; k += 4) {
    const int dA = k + (hs << 1);
    v2f a;  a.x  = arow[dA];                  a.y  = arow[dA + 1];
    v2f b0; b0.x = Wl[dA * kYp + l16];        b0.y = Wl[(dA + 1) * kYp + l16];
    v2f b1; b1.x = Wl[dA * kYp + 16 + l16];   b1.y = Wl[(dA + 1) * kYp + 16 + l16];
    c0 = __builtin_amdgcn_wmma_f32_16x16x4_f32(false, a, false, b0, (short)0, c0,
                                               false, false);
    c1 = __builtin_amdgcn_wmma_f32_16x16x4_f32(false, a, false, b1, (short)0, c1,
                                               false, false);
  }

#pragma unroll
  for (int r = 0; r < 8; ++r) {
    const size_t rowo = (size_t)(row0 + r + (hs << 3));
    emit[rowo * kY + l16] = c0[r];
    if (l16 < kY - 16) emit[rowo * kY + 16 + l16] = c1[r];
  }
}

// ---------------------------------------------------------------------------
// Kernel 2: one wave32 per chain. Lane = destination label y1 (26 active).
// Forward: l_i[y1] = max_{y'} (e_{i-1}[y'] + T[y', y1] + l_{i-1}[y']),
// broadcast of m[y'] = e+l via v_readlane; argmax via tournament tree with
// ties kept by the lower slot (not always the first index, unlike jnp.argmax). Backpointers in LDS.
// Backtrace: dependent ds_load_u8 chain; one-hot rows written per step.
// NOTE: `emit` aliases `out` — all emit reads for a chain finish (forward
// pass) before that chain's output rows are written (backtrace).
// ---------------------------------------------------------------------------
__global__ __launch_bounds__(32) void crf_viterbi(
    const float* emit, const float* __restrict__ T, float* out) {
  extern __shared__ unsigned char smem[];
  unsigned char* bp = smem;                          // [kS][kY] backpointers
  float* fin = (float*)(smem + kS * kY);             // kS*kY = 53248 (16B-aligned)

  const int lane  = threadIdx.x;
  const int lc    = (lane < kY) ? lane : (kY - 1);   // clamp for pad lanes
  const int chain = blockIdx.x;
  const float* em = emit + (size_t)chain * kS * kY;

  // T column for this destination label: tcol[j] = T[j][y1]
  float tcol[kY];
#pragma unroll
  for (int j = 0; j < kY; ++j) tcol[j] = T[j * kY + lc];

  float l     = 0.0f;        // lookup[i-1][lane]
  float e_im1 = em[lc];      // emit row 0

  for (int i = 1; i < kS; ++i) {
    const float e_i = em[i * kY + lc];                       // row i (next step)
    __builtin_prefetch(&em[(i + 24) * kY + lc], 0, 0);       // stream ahead

    const float m = e_im1 + l;                               // lane = source y'
    float val[32];
    int   idx[32];
#pragma unroll
    for (int j = 0; j < kY; ++j) {
      val[j] = __int_as_float(
                   __builtin_amdgcn_readlane(__float_as_int(m), j)) + tcol[j];
      idx[j] = j;
    }
#pragma unroll
    for (int j = kY; j < 32; ++j) { val[j] = -3.0e38f; idx[j] = 0; }

    // Tournament argmax, lower index wins ties (>= keeps slot j).
#pragma unroll
    for (int w = 16; w >= 1; w >>= 1) {
#pragma unroll
      for (int j = 0; j < 16; ++j) {
        if (j < w) {
          const bool c = val[j] >= val[j + w];
          val[j] = c ? val[j] : val[j + w];
          idx[j] = c ? idx[j] : idx[j + w];
        }
      }
    }
    l = val[0];
    if (lane < kY) bp[i * kY + lane] = (unsigned char)idx[0];
    e_im1 = e_i;
  }

  // last = argmax(em[S-1] + lookup[S-1]) across lanes, first-index-wins.
  fin[lane] = (lane < kY) ? (e_im1 + l) : -3.0e38f;
  __syncthreads();
  int cur = 0;
  float best = fin[0];
#pragma unroll
  for (int j = 1; j < kY; ++j) {
    if (fin[j] > best) { best = fin[j]; cur = j; }
  }

  // Backtrace + one-hot output.
  float* orow = out + (size_t)chain * kS * kY;
  for (int i = kS - 1; i >= 1; --i) {
    if (lane < kY) orow[(size_t)i * kY + lane] = (lane == cur) ? 1.0f : 0.0f;
    cur = bp[i * kY + cur];                    // broadcast LDS read
  }
  if (lane < kY) orow[lane] = (lane == cur) ? 1.0f : 0.0f;
}

// ---------------------------------------------------------------------------
extern "C" void kernel_launch(void* const* d_in, const int* in_sizes, int n_in,
                              void* d_out, int out_size, void* d_ws,
                              size_t ws_size, hipStream_t stream) {
  (void)in_sizes; (void)n_in; (void)d_ws; (void)ws_size; (void)out_size;
  const float* X = (const float*)d_in[0];   // [B, S, D]
  const float* W = (const float*)d_in[1];   // [D, Y]
  const float* T = (const float*)d_in[2];   // [Y, Y]
  float* out = (float*)d_out;               // [B, S, Y] one-hot fp32

  // emit scratch lives in d_out ([B*S,26] fp32 == output size); kernel 2
  // fully overwrites it with the one-hot path, so every call is identical.
  crf_emit_wmma<<<(kB * kS) / (16 * 8), 256, 0, stream>>>(X, W, out);

  const size_t smem = (size_t)kS * kY + 32 * sizeof(float);  // ~53.4 KB
  crf_viterbi<<<kB, 32, smem, stream>>>(out, T, out);
}